// NodeModel_12953621365293
// MI455X (gfx1250) — compile-verified
//
#include <hip/hip_runtime.h>

#define N_NODES 50000
#define N_EDGES 1250000
#define HF 64

typedef __attribute__((ext_vector_type(2))) float v2f;
typedef __attribute__((ext_vector_type(8))) float v8f;

// ---------------------------------------------------------------------------
// Kernel 1: initialize combined = [x | 0]  (combined lives inside d_out)
// ---------------------------------------------------------------------------
__global__ void init_combined_kernel(const float* __restrict__ x,
                                     float* __restrict__ combined) {
    int tid = blockIdx.x * blockDim.x + threadIdx.x;
    const int total = N_NODES * 128;
    if (tid >= total) return;
    int n = tid >> 7;       // node
    int c = tid & 127;      // column in combined
    combined[tid] = (c < HF) ? x[n * HF + c] : 0.0f;
}

// ---------------------------------------------------------------------------
// Kernel 2: scatter-add edge_attr into combined[:, 64:128] with f32 atomics.
// One thread = one (edge, 4-feature chunk); float4 vectorized source read.
// Destination (25.6 MB) is L2-resident on MI455X (192 MB L2).
// ---------------------------------------------------------------------------
__global__ void scatter_kernel(const int* __restrict__ row,
                               const float* __restrict__ edge_attr,
                               float* __restrict__ combined) {
    long long tid = (long long)blockIdx.x * blockDim.x + threadIdx.x;
    long long e  = tid >> 4;          // edge id
    int chunk    = (int)(tid & 15);   // 16 chunks of 4 floats = 64 features
    if (e >= N_EDGES) return;
    int r = row[e];
    float4 v = ((const float4*)(edge_attr + e * HF))[chunk];
    float* dst = combined + (long long)r * 128 + 64 + chunk * 4;
    unsafeAtomicAdd(dst + 0, v.x);   // global_atomic_add_f32 (no CAS loop)
    unsafeAtomicAdd(dst + 1, v.y);
    unsafeAtomicAdd(dst + 2, v.z);
    unsafeAtomicAdd(dst + 3, v.w);
}

// ---------------------------------------------------------------------------
// Kernel 3: out = silu(combined @ W1 + b1) @ W2 + b2 via V_WMMA_F32_16X16X4_F32
// One wave32 owns a 16-node tile (M=16), producing all 64 outputs (4 N-tiles).
// W1/W2 staged in LDS pair-interleaved so B operands are single ds_load_b64.
// Hidden tile staged per-wave in LDS to convert D-layout -> A-layout.
// ---------------------------------------------------------------------------
#define WAVES_PER_BLOCK 8

__device__ __forceinline__ float fast_silu(float v) {
    // v * sigmoid(v); v_exp_f32 + v_rcp_f32 (~1 ulp), avoids the precise
    // div_scale/div_fmas expansion which costs ~8 VALU ops per element.
    return v * __builtin_amdgcn_rcpf(1.0f + __expf(-v));
}

__launch_bounds__(256)
__global__ void mlp_wmma_kernel(const float* __restrict__ combined,
                                const float* __restrict__ W1,
                                const float* __restrict__ b1,
                                const float* __restrict__ W2,
                                const float* __restrict__ b2,
                                float* __restrict__ out) {
    // Pair-interleaved weights: s[(k>>1)*128 + n*2 + (k&1)] = W[k][n]
    __shared__ float sW1[128 * 64];                 // 32 KB
    __shared__ float sW2[64 * 64];                  // 16 KB
    __shared__ float sH[WAVES_PER_BLOCK * 16 * 64]; // 32 KB (per-wave h tiles)

    const int tid = threadIdx.x;
    for (int i = tid; i < 128 * 64; i += 256) {
        int k = i >> 6, n = i & 63;
        sW1[(k >> 1) * 128 + n * 2 + (k & 1)] = W1[i];
    }
    for (int i = tid; i < 64 * 64; i += 256) {
        int k = i >> 6, n = i & 63;
        sW2[(k >> 1) * 128 + n * 2 + (k & 1)] = W2[i];
    }
    __syncthreads();

    const int wave = tid >> 5;
    const int lane = tid & 31;
    const int tile = blockIdx.x * WAVES_PER_BLOCK + wave;
    if (tile >= N_NODES / 16) return;           // wave-uniform; EXEC stays full
    const int base = tile * 16;

    const int m     = lane & 15;                // A-operand: lane -> matrix row M
    const int nlo   = lane & 15;                // B/C/D: lane -> column N (within N-tile)
    const int khalf = (lane >> 4) << 1;         // A/B: hi half-wave covers K+2,K+3
    const int hi8   = (lane >> 4) << 3;         // C/D: hi half-wave covers rows M+8

    // ---- GEMM1: [16x128] @ [128x64], C initialized with b1 ----
    v8f acc[4];
    #pragma unroll
    for (int nt = 0; nt < 4; ++nt) {
        float bv = b1[nt * 16 + nlo];
        #pragma unroll
        for (int r = 0; r < 8; ++r) acc[nt][r] = bv;
    }
    const float2* crow = (const float2*)(combined + (long long)(base + m) * 128);
    for (int k = 0; k < 128; k += 4) {
        float2 af = crow[(k + khalf) >> 1];     // global_load_b64: K pair for this lane
        v2f a; a.x = af.x; a.y = af.y;
        const float* w1p = sW1 + ((k + khalf) >> 1) * 128;
        #pragma unroll
        for (int nt = 0; nt < 4; ++nt) {
            float2 bf = *(const float2*)(w1p + (nt * 16 + nlo) * 2);  // ds_load_b64
            v2f b; b.x = bf.x; b.y = bf.y;
            acc[nt] = __builtin_amdgcn_wmma_f32_16x16x4_f32(
                false, a, false, b, (short)0, acc[nt], false, false);
        }
    }

    // ---- SiLU, stash hidden tile row-major in per-wave LDS (no barrier: same wave) ----
    float* hbase = sH + wave * (16 * 64);
    #pragma unroll
    for (int nt = 0; nt < 4; ++nt) {
        #pragma unroll
        for (int r = 0; r < 8; ++r) {
            hbase[(r + hi8) * 64 + nt * 16 + nlo] = fast_silu(acc[nt][r]);
        }
    }

    // ---- GEMM2: [16x64] @ [64x64], C initialized with b2 ----
    v8f acc2[4];
    #pragma unroll
    for (int nt = 0; nt < 4; ++nt) {
        float bv = b2[nt * 16 + nlo];
        #pragma unroll
        for (int r = 0; r < 8; ++r) acc2[nt][r] = bv;
    }
    const float2* hrow = (const float2*)(hbase + m * 64);
    for (int k = 0; k < 64; k += 4) {
        float2 af = hrow[(k + khalf) >> 1];     // ds_load_b64
        v2f a; a.x = af.x; a.y = af.y;
        const float* w2p = sW2 + ((k + khalf) >> 1) * 128;
        #pragma unroll
        for (int nt = 0; nt < 4; ++nt) {
            float2 bf = *(const float2*)(w2p + (nt * 16 + nlo) * 2);  // ds_load_b64
            v2f b; b.x = bf.x; b.y = bf.y;
            acc2[nt] = __builtin_amdgcn_wmma_f32_16x16x4_f32(
                false, a, false, b, (short)0, acc2[nt], false, false);
        }
    }

    // ---- store D (16x64) to out ----
    #pragma unroll
    for (int nt = 0; nt < 4; ++nt) {
        #pragma unroll
        for (int r = 0; r < 8; ++r) {
            out[(long long)(base + r + hi8) * 64 + nt * 16 + nlo] = acc2[nt][r];
        }
    }
}

// ---------------------------------------------------------------------------
extern "C" void kernel_launch(void* const* d_in, const int* in_sizes, int n_in,
                              void* d_out, int out_size, void* d_ws, size_t ws_size,
                              hipStream_t stream) {
    const int*   edge_index = (const int*)d_in[0];   // [2, N_EDGES]; row = first N_EDGES
    const float* edge_attr  = (const float*)d_in[1]; // [N_EDGES, 64]
    const float* x          = (const float*)d_in[2]; // [N_NODES, 64]
    const float* W1         = (const float*)d_in[3]; // [128, 64]
    const float* b1         = (const float*)d_in[4]; // [64]
    const float* W2         = (const float*)d_in[5]; // [64, 64]
    const float* b2         = (const float*)d_in[6]; // [64]

    float* out      = (float*)d_out;                    // [N_NODES, 64]
    float* combined = out + (long long)N_NODES * HF;    // [N_NODES, 128] (2nd output)

    {   // combined = [x | 0]
        int total = N_NODES * 128;
        init_combined_kernel<<<(total + 255) / 256, 256, 0, stream>>>(x, combined);
    }
    {   // combined[:, 64:] += scatter(edge_attr by row)
        long long total = (long long)N_EDGES * 16;
        scatter_kernel<<<(int)((total + 255) / 256), 256, 0, stream>>>(
            edge_index, edge_attr, combined);
    }
    {   // out = silu(combined@W1+b1)@W2+b2
        int tiles  = N_NODES / 16;                 // 3125 exact tiles
        int blocks = (tiles + WAVES_PER_BLOCK - 1) / WAVES_PER_BLOCK;
        mlp_wmma_kernel<<<blocks, 256, 0, stream>>>(combined, W1, b1, W2, b2, out);
    }
}